// AttentionTransformer_Inspired_60636348285214
// MI455X (gfx1250) — compile-verified
//
#include <hip/hip_runtime.h>
#include <hip/hip_bf16.h>

#define BN_EPS 1e-3f

typedef float v2f __attribute__((ext_vector_type(2)));
typedef float v8f __attribute__((ext_vector_type(8)));

static __device__ __forceinline__ float wave_red_add(float v) {
#pragma unroll
  for (int off = 16; off > 0; off >>= 1) v += __shfl_xor(v, off, 32);
  return v;
}
static __device__ __forceinline__ float wave_red_max(float v) {
#pragma unroll
  for (int off = 16; off > 0; off >>= 1) v = fmaxf(v, __shfl_xor(v, off, 32));
  return v;
}

// Tile configuration
constexpr int Dm  = 1024;   // model dim (K and N)
constexpr int BM  = 32;     // output rows per workgroup
constexpr int KC  = 64;     // K chunk staged in LDS
constexpr int LDA = 68;     // padded dword stride of A chunk (bank-conflict free)
constexpr int LDC = 1032;   // padded dword stride of C tile

// smem layout (dwords): sc[1024] | sh[1024] | A[BM*LDA] | C[BM*LDC]
constexpr int SMEM_DWORDS = 1024 + 1024 + BM * LDA + BM * LDC;  // 37248 -> 148992 B

__global__ __launch_bounds__(256) void tabnet_sparsemax_fused(
    const float* __restrict__ inp,     // [B, 1024]
    const float* __restrict__ priors,  // [B, 1024]
    const float* __restrict__ W,       // [1024, 1024] row-major (K x N)
    const float* __restrict__ gamma,
    const float* __restrict__ beta,
    const float* __restrict__ mmean,
    const float* __restrict__ mvar,
    float* __restrict__ out) {
  extern __shared__ float smem[];
  float* sc = smem;              // BN scale per column
  float* sh = sc + Dm;           // BN shift per column
  float* lA = sh + Dm;           // A chunk  [BM][LDA]
  float* lC = lA + BM * LDA;     // C tile   [BM][LDC]

  const int tid  = threadIdx.x;
  const int wave = tid >> 5;
  const int lane = tid & 31;
  const int lo16 = lane & 15;        // M index within 16 (A), N index within 16 (B/C)
  const int half = lane >> 4;        // 0: lanes 0-15, 1: lanes 16-31
  const int koff = half << 1;        // K-pair selector per ISA 16x4 / 4x16 layouts
  const int row0 = blockIdx.x * BM;
  const int n0   = wave * 128;       // each wave owns 128 output columns

  // ---- Fold BatchNorm into per-column scale/shift ----
  for (int c = tid; c < Dm; c += 256) {
    const float s = gamma[c] * rsqrtf(mvar[c] + BN_EPS);
    sc[c] = s;
    sh[c] = beta[c] - mmean[c] * s;
  }
  __syncthreads();

  // ---- GEMM: acc[mt][nt] = 16x16 f32 tile via V_WMMA_F32_16X16X4_F32 ----
  v8f acc[2][8];
#pragma unroll
  for (int mt = 0; mt < 2; ++mt)
#pragma unroll
    for (int nt = 0; nt < 8; ++nt)
#pragma unroll
      for (int e = 0; e < 8; ++e) acc[mt][nt][e] = 0.0f;

  for (int kc = 0; kc < Dm; kc += KC) {
    // Cooperative A-chunk load: 32 rows x 64 K, 8 f32 per thread (2x b128)
    {
      const int r = tid >> 3;
      const int c = (tid & 7) * 8;
      const float* gp = inp + (size_t)(row0 + r) * Dm + kc + c;
      const float4 p0 = ((const float4*)gp)[0];
      const float4 p1 = ((const float4*)gp)[1];
      float4* dp = (float4*)&lA[r * LDA + c];
      dp[0] = p0;
      dp[1] = p1;
    }
    __syncthreads();

    const float* Wp0 = W + (size_t)kc * Dm + n0 + lo16;
#pragma unroll 4
    for (int k = 0; k < KC; k += 4) {
      // A fragments (16x4): lane holds row lo16, K = k+koff, k+koff+1
      v2f a0, a1;
      {
        const float* ap0 = &lA[lo16 * LDA + k + koff];
        const float* ap1 = &lA[(16 + lo16) * LDA + k + koff];
        a0.x = ap0[0]; a0.y = ap0[1];
        a1.x = ap1[0]; a1.y = ap1[1];
      }
      // B fragments (4x16): VGPR0 = W[k+koff][col], VGPR1 = W[k+koff+1][col]
      const float* wp = Wp0 + (size_t)(k + koff) * Dm;
#pragma unroll
      for (int nt = 0; nt < 8; ++nt) {
        v2f b;
        b.x = wp[nt * 16];
        b.y = wp[Dm + nt * 16];
        acc[0][nt] = __builtin_amdgcn_wmma_f32_16x16x4_f32(
            false, a0, false, b, (short)0, acc[0][nt], false, false);
        acc[1][nt] = __builtin_amdgcn_wmma_f32_16x16x4_f32(
            false, a1, false, b, (short)0, acc[1][nt], false, false);
      }
    }
    __syncthreads();
  }

  // ---- Epilogue: BN + prior mask, scatter C tile into LDS ----
#pragma unroll
  for (int nt = 0; nt < 8; ++nt) {
    const int col  = n0 + nt * 16 + lo16;
    const float s0 = sc[col];
    const float s1 = sh[col];
#pragma unroll
    for (int mt = 0; mt < 2; ++mt) {
#pragma unroll
      for (int v = 0; v < 8; ++v) {
        const int row = mt * 16 + v + 8 * half;  // C/D layout: VGPR v -> M = v (+8 hi half)
        float val = acc[mt][nt][v] * s0 + s1;
        val *= priors[(size_t)(row0 + row) * Dm + col];
        lC[row * LDC + col] = val;
      }
    }
  }
  __syncthreads();

  // ---- Sparsemax per row (4 rows per wave): bisection + exact refinement ----
#pragma unroll
  for (int rr = 0; rr < 4; ++rr) {
    const int r = wave * 4 + rr;
    float z[32];
#pragma unroll
    for (int j = 0; j < 32; ++j) z[j] = lC[r * LDC + j * 32 + lane];

    float m = z[0];
#pragma unroll
    for (int j = 1; j < 32; ++j) m = fmaxf(m, z[j]);
    m = wave_red_max(m);

    // f(tau) = sum(max(z - tau, 0)) is monotone decreasing; tau* in [m-1, m]
    float tlo = m - 1.0f, thi = m;
    for (int it = 0; it < 30; ++it) {
      const float mid = 0.5f * (tlo + thi);
      float s = 0.0f;
#pragma unroll
      for (int j = 0; j < 32; ++j) s += fmaxf(z[j] - mid, 0.0f);
      s = wave_red_add(s);
      if (s >= 1.0f) tlo = mid; else thi = mid;
    }
    // Exact tau from the recovered support: tau = (sum_{z>tlo} z - 1) / k
    float cnt = 0.0f, ssum = 0.0f;
#pragma unroll
    for (int j = 0; j < 32; ++j) {
      if (z[j] > tlo) { cnt += 1.0f; ssum += z[j]; }
    }
    cnt  = wave_red_add(cnt);
    ssum = wave_red_add(ssum);
    const float tau = (ssum - 1.0f) / cnt;

    float* op = out + (size_t)(row0 + r) * Dm;
#pragma unroll
    for (int j = 0; j < 32; ++j) op[j * 32 + lane] = fmaxf(z[j] - tau, 0.0f);
  }
}

extern "C" void kernel_launch(void* const* d_in, const int* in_sizes, int n_in,
                              void* d_out, int out_size, void* d_ws, size_t ws_size,
                              hipStream_t stream) {
  (void)n_in; (void)d_ws; (void)ws_size; (void)out_size;
  const float* inp    = (const float*)d_in[0];
  const float* priors = (const float*)d_in[1];
  const float* W      = (const float*)d_in[2];
  const float* gamma  = (const float*)d_in[3];
  const float* beta   = (const float*)d_in[4];
  const float* mmean  = (const float*)d_in[5];
  const float* mvar   = (const float*)d_in[6];
  float* out = (float*)d_out;

  const int B = in_sizes[0] / Dm;  // 65536
  const size_t smem_bytes = (size_t)SMEM_DWORDS * sizeof(float);  // ~149 KB (CDNA5: 320 KB/WGP)

  // Opt in to >64KB dynamic LDS (no-op / harmless if already permitted).
  (void)hipFuncSetAttribute((const void*)tabnet_sparsemax_fused,
                            hipFuncAttributeMaxDynamicSharedMemorySize,
                            (int)smem_bytes);

  dim3 grid(B / BM), block(256);
  tabnet_sparsemax_fused<<<grid, block, smem_bytes, stream>>>(
      inp, priors, W, gamma, beta, mmean, mvar, out);
}